// DotAttentionV2_72378788872875
// MI455X (gfx1250) — compile-verified
//
#include <hip/hip_runtime.h>
#include <math.h>

// DotAttentionV2: B=256,H=1,T=32,F=64,E=128, fp32.
// Memory-bound (AI ~0.33 FLOP/B): stream q,k,v once; WMMA f32 16x16x4 for alpha@V.

typedef float v2f __attribute__((ext_vector_type(2)));
typedef float v8f __attribute__((ext_vector_type(8)));

#define B_ 256
#define T_ 32
#define F_ 64
#define E_ 128

__global__ __launch_bounds__(256) void dotattn_v2_kernel(
    const float* __restrict__ q,   // (B,1,T,F,E)
    const float* __restrict__ k,   // (B,1,F,T,E)
    const float* __restrict__ v,   // (B,1,F,T,E)
    float* __restrict__ out)       // (B,T,E)
{
    __shared__ __align__(16) float alphaS[8][F_];   // per-wave alpha (2 KB)

    const int lane = threadIdx.x & 31;
    const int wave = threadIdx.x >> 5;
    const int bt   = blockIdx.x * 8 + wave;  // grid sized exactly: bt < B*T
    const int b    = bt >> 5;                // bt / T_
    const int t    = bt & (T_ - 1);

    // ---------------- Phase 1: scores s[f] = dot(q[f,:], k[f,:]) / sqrt(E)
    // lane owns rows f=lane and f=lane+32; float4 streaming loads.
    const size_t qbase = (size_t)bt * (size_t)(F_ * E_);
    const size_t kbase = (size_t)b * (size_t)(F_ * T_ * E_) + (size_t)t * E_;

    const float4* q0 = (const float4*)(q + qbase + (size_t)lane * E_);
    const float4* q1 = (const float4*)(q + qbase + (size_t)(lane + 32) * E_);
    const float4* k0 = (const float4*)(k + kbase + (size_t)lane * (T_ * E_));
    const float4* k1 = (const float4*)(k + kbase + (size_t)(lane + 32) * (T_ * E_));

    float s0 = 0.f, s1 = 0.f;
#pragma unroll 8
    for (int i = 0; i < E_ / 4; ++i) {
        float4 a = q0[i], c = k0[i];
        s0 += a.x * c.x + a.y * c.y + a.z * c.z + a.w * c.w;
        float4 d = q1[i], e = k1[i];
        s1 += d.x * e.x + d.y * e.y + d.z * e.z + d.w * e.w;
    }
    const float scale = 0.08838834764831845f;  // 1/sqrt(128)
    s0 *= scale; s1 *= scale;

    // ---------------- Phase 2: softmax over 64 scores (wave32 allreduce)
    float m = fmaxf(s0, s1);
    for (int off = 16; off; off >>= 1) m = fmaxf(m, __shfl_xor(m, off, 32));
    float e0 = __expf(s0 - m), e1 = __expf(s1 - m);
    float sum = e0 + e1;
    for (int off = 16; off; off >>= 1) sum += __shfl_xor(sum, off, 32);
    const float inv = 1.0f / sum;

    alphaS[wave][lane]      = e0 * inv;   // f = lane
    alphaS[wave][lane + 32] = e1 * inv;   // f = lane + 32
    __syncthreads();

    // ---------------- Phase 3: out[e] = sum_f alpha[f] * v[f,e] via WMMA f32 16x16x4
    // A(16x4) layout: lane l<16 holds A[l,K0],A[l,K1]; lane>=16 holds A[l-16,K2],A[l-16,K3].
    // Only A row 0 (lanes 0 and 16) reaches D row 0, which is all we store —
    // so EVERY lane loads the same alpha pair (LDS broadcast, zero divergence);
    // D rows 1..15 hold harmless finite garbage that is never read.
    const float* vbase = v + (size_t)b * (size_t)(F_ * T_ * E_) + (size_t)t * E_;
    const int n  = lane & 15;
    const int hi = lane >> 4;              // 0: carries K0/K1, 1: carries K2/K3

    // Hoist all 16 A-fragments (K-steps) out of the e-tile loop: 16x ds_load_b64.
    const v2f* aP = (const v2f*)(alphaS[wave]);
    v2f Af[16];
#pragma unroll
    for (int j = 0; j < 16; ++j) Af[j] = aP[2 * j + hi];

    float* obase = out + (size_t)bt * E_;

#pragma unroll
    for (int nt = 0; nt < 8; ++nt) {               // e-tiles of 16
        v8f c = {};
        const int ebase = nt * 16 + n;
#pragma unroll
        for (int j = 0; j < 16; ++j) {             // K-steps over f, 4 at a time
            const int fr = 4 * j + 2 * hi;
            v2f Bm;
            Bm.x = vbase[(size_t)fr * (T_ * E_) + ebase];
            Bm.y = vbase[(size_t)(fr + 1) * (T_ * E_) + ebase];
            c = __builtin_amdgcn_wmma_f32_16x16x4_f32(
                    false, Af[j], false, Bm, (short)0, c, false, false);
        }
        if (!hi) obase[nt * 16 + n] = c[0];        // D[0, n]
    }
}

extern "C" void kernel_launch(void* const* d_in, const int* in_sizes, int n_in,
                              void* d_out, int out_size, void* d_ws, size_t ws_size,
                              hipStream_t stream) {
    const float* q = (const float*)d_in[0];
    const float* k = (const float*)d_in[1];
    const float* v = (const float*)d_in[2];
    float* out = (float*)d_out;
    dim3 grid((B_ * T_) / 8);   // 1024 blocks, 8 waves each, 1 wave per (b,t)
    dim3 block(256);
    hipLaunchKernelGGL(dotattn_v2_kernel, grid, block, 0, stream, q, k, v, out);
}